// DSRQSLoss_31894427140770
// MI455X (gfx1250) — compile-verified
//
#include <hip/hip_runtime.h>
#include <hip/hip_bf16.h>

// MI455X (gfx1250) implementation of DSRQS loss.
// Bandwidth-bound streaming pass (402MB -> ~17us floor at 23.3TB/s) with
// sorted-qid run compaction + LDS atomics; streamed inputs use non-temporal
// loads (read-once, > L2 capacity) so the 1.6MB group-accumulator working set
// stays L2-resident. Group reduction uses V_WMMA_F32_16X16X4_F32 as a 64-way
// register-resident sum engine; f64 scalar accumulators for accuracy.

#define NG 100000
#define LAM 0.5
#define GAMMA 0.2f

constexpr int TPB   = 256;   // 8 wave32 per block
constexpr int VPT   = 16;    // elements per thread (4x b128 loads per array)
constexpr int CHUNK = TPB * VPT;  // 4096 contiguous elements per block
constexpr int SLOTS = 512;   // LDS qid window (expected span ~12 per chunk)

typedef __attribute__((ext_vector_type(2))) float v2f;
typedef __attribute__((ext_vector_type(4))) float v4f;
typedef __attribute__((ext_vector_type(8))) float v8f;
typedef __attribute__((ext_vector_type(4))) int   v4i;

__global__ void k_zero(float* __restrict__ p, int n) {
  int i = blockIdx.x * blockDim.x + threadIdx.x;
  if (i < n) p[i] = 0.0f;
}

__global__ __launch_bounds__(TPB) void k_main(
    const float* __restrict__ scores, const int* __restrict__ labels,
    const int* __restrict__ qids, double* __restrict__ dacc,
    float* __restrict__ pcnt, float* __restrict__ ncnt,
    float* __restrict__ psum, float* __restrict__ nsum, long long n) {
  __shared__ float slot[SLOTS][4];
  __shared__ float s_bce;
  for (int i = threadIdx.x; i < SLOTS; i += TPB) {
    slot[i][0] = 0.f; slot[i][1] = 0.f; slot[i][2] = 0.f; slot[i][3] = 0.f;
  }
  if (threadIdx.x == 0) s_bce = 0.f;
  __syncthreads();

  const long long blockStart = (long long)blockIdx.x * CHUNK;
  const int qfirst = qids[blockStart];  // uniform (sorted qids)
  const long long tbase = blockStart + (long long)threadIdx.x * VPT;

  float bce = 0.f;

  auto flush = [&](int q, float b0, float b1, float b2, float b3) {
    int sl = q - qfirst;
    if (sl >= 0 && sl < SLOTS) {               // LDS fast path (ds_add_f32)
      atomicAdd(&slot[sl][0], b0); atomicAdd(&slot[sl][1], b1);
      atomicAdd(&slot[sl][2], b2); atomicAdd(&slot[sl][3], b3);
    } else if (q >= 0 && q < NG) {             // rare fallback
      atomicAdd(&pcnt[q], b0); atomicAdd(&ncnt[q], b1);
      atomicAdd(&psum[q], b2); atomicAdd(&nsum[q], b3);
    }
  };

  if (tbase < n) {
    float sv[VPT]; int yv[VPT]; int qv[VPT];
    int cnt;
    if (tbase + VPT <= n) {  // full chunk: 16B-aligned b128 non-temporal loads
      cnt = VPT;
      const v4f* S4 = reinterpret_cast<const v4f*>(scores + tbase);
      const v4i* L4 = reinterpret_cast<const v4i*>(labels + tbase);
      const v4i* Q4 = reinterpret_cast<const v4i*>(qids + tbase);
#pragma unroll
      for (int j = 0; j < VPT / 4; ++j) {
        v4f s = __builtin_nontemporal_load(S4 + j);
        v4i l = __builtin_nontemporal_load(L4 + j);
        v4i q = __builtin_nontemporal_load(Q4 + j);
#pragma unroll
        for (int k = 0; k < 4; ++k) {
          sv[4 * j + k] = s[k];
          yv[4 * j + k] = l[k];
          qv[4 * j + k] = q[k];
        }
      }
    } else {                 // tail
      cnt = (int)(n - tbase);
#pragma unroll
      for (int e = 0; e < VPT; ++e) {
        if (e < cnt) { sv[e]=scores[tbase+e]; yv[e]=labels[tbase+e]; qv[e]=qids[tbase+e]; }
        else         { sv[e]=0.5f; yv[e]=0; qv[e]=-1; }
      }
    }

    int cur = qv[0];
    float a0 = 0.f, a1 = 0.f, a2 = 0.f, a3 = 0.f;
#pragma unroll
    for (int e = 0; e < VPT; ++e) {
      if (e < cnt) {
        float s = sv[e]; int yi = yv[e]; int q = qv[e];
        float y = (float)yi;
        bce -= __logf(yi ? s : 1.f - s);       // native v_log_f32 path
        if (q != cur) {
          flush(cur, a0, a1, a2, a3);
          cur = q; a0 = a1 = a2 = a3 = 0.f;
        }
        a0 += y; a1 += 1.f - y; a2 += s * y; a3 += s - s * y;
      }
    }
    flush(cur, a0, a1, a2, a3);
  }

  // wave32 reduce BCE, then block combine, one f64 atomic per block
#pragma unroll
  for (int o = 16; o > 0; o >>= 1) bce += __shfl_xor(bce, o, 32);
  if ((threadIdx.x & 31) == 0) atomicAdd(&s_bce, bce);
  __syncthreads();
  if (threadIdx.x == 0) atomicAdd(&dacc[0], (double)s_bce);

  // flush LDS group window to global accumulators (L2-resident, regular TH)
  for (int i = threadIdx.x; i < SLOTS; i += TPB) {
    float b0 = slot[i][0], b1 = slot[i][1];
    if (b0 != 0.f || b1 != 0.f) {              // any element landed here
      int g = qfirst + i;
      if (g < NG) {
        atomicAdd(&pcnt[g], b0); atomicAdd(&ncnt[g], b1);
        atomicAdd(&psum[g], slot[i][2]); atomicAdd(&nsum[g], slot[i][3]);
      }
    }
  }
}

// Group reduction via V_WMMA_F32_16X16X4_F32 used as a 64-way sum engine:
// D = A * ones(4x16) + C  =>  each D column accumulates the total of A.
__global__ __launch_bounds__(TPB) void k_group(
    double* __restrict__ dacc, const float* __restrict__ pcnt,
    const float* __restrict__ ncnt, const float* __restrict__ psum,
    const float* __restrict__ nsum, int ng) {
  const int lane   = threadIdx.x & 31;
  const int wave   = (int)((blockIdx.x * blockDim.x + threadIdx.x) >> 5);
  const int nwaves = (int)((gridDim.x * blockDim.x) >> 5);
  const int per    = nwaves * 64;
  const int iters  = (ng + per - 1) / per;     // uniform trip count (EXEC=all at WMMA)

  v8f ch = {};                                  // hinge accumulator (16x16 f32 C/D)
  v8f cv = {};                                  // valid-count accumulator
  v2f ones; ones.x = 1.f; ones.y = 1.f;         // B = all-ones 4x16

  for (int it = 0; it < iters; ++it) {
    int base = (it * nwaves + wave) * 64;
    v2f ah; ah.x = 0.f; ah.y = 0.f;
    v2f av; av.x = 0.f; av.y = 0.f;
#pragma unroll
    for (int half = 0; half < 2; ++half) {
      int g = base + half * 32 + lane;
      float h = 0.f, v = 0.f;
      if (g < ng) {
        float pc = pcnt[g], nc = ncnt[g];
        if (pc > 0.f && nc > 0.f) {
          // Native v_rcp_f32 (~1 ulp): pc/nc are small positive integer-valued
          // floats feeding a tolerance-scale loss scalar, so skip the IEEE
          // div_scale/div_fixup sequence the compiler otherwise emits.
          float delta = psum[g] * __builtin_amdgcn_rcpf(pc)
                      - nsum[g] * __builtin_amdgcn_rcpf(nc);
          h = fmaxf(GAMMA - delta, 0.f);
          v = 1.f;
        }
      }
      if (half == 0) { ah.x = h; av.x = v; } else { ah.y = h; av.y = v; }
    }
    // (neg_a, A, neg_b, B, c_mod, C, reuse_a, reuse_b)
    ch = __builtin_amdgcn_wmma_f32_16x16x4_f32(false, ah, false, ones,
                                               (short)0, ch, false, false);
    cv = __builtin_amdgcn_wmma_f32_16x16x4_f32(false, av, false, ones,
                                               (short)0, cv, false, false);
  }

  float sh = ch[0]+ch[1]+ch[2]+ch[3]+ch[4]+ch[5]+ch[6]+ch[7];
  float sv = cv[0]+cv[1]+cv[2]+cv[3]+cv[4]+cv[5]+cv[6]+cv[7];
  // lane L holds D[0..7][L], lane L+16 holds D[8..15][L]: pair them
  sh += __shfl_xor(sh, 16, 32);
  sv += __shfl_xor(sv, 16, 32);
  if (lane == 0) {
    atomicAdd(&dacc[1], (double)sh);
    atomicAdd(&dacc[2], (double)sv);
  }
}

__global__ void k_final(const double* __restrict__ dacc, float* __restrict__ out,
                        long long n) {
  double lce = dacc[0] / (double)n;
  double nv  = dacc[2];
  double ldc = (nv > 0.0) ? (dacc[1] / nv) : 0.0;
  out[0] = (float)(lce + LAM * ldc);
}

extern "C" void kernel_launch(void* const* d_in, const int* in_sizes, int n_in,
                              void* d_out, int out_size, void* d_ws, size_t ws_size,
                              hipStream_t stream) {
  const float* scores = (const float*)d_in[0];
  const int*   labels = (const int*)d_in[1];
  const int*   qids   = (const int*)d_in[2];
  long long n = (long long)in_sizes[0];

  // ws layout: [0,24) three f64 accumulators (bce, hinge, valid); pad to 32B;
  // then 4*NG f32 group arrays.
  double* dacc = (double*)d_ws;
  float*  grp  = (float*)d_ws + 8;
  float *pcnt = grp, *ncnt = grp + NG, *psum = grp + 2 * NG, *nsum = grp + 3 * NG;

  int zn = 8 + 4 * NG;
  k_zero<<<(zn + 255) / 256, 256, 0, stream>>>((float*)d_ws, zn);

  long long blocks = (n + CHUNK - 1) / CHUNK;   // 8192 for N=2^25
  k_main<<<dim3((unsigned)blocks), TPB, 0, stream>>>(scores, labels, qids,
                                                     dacc, pcnt, ncnt, psum, nsum, n);

  k_group<<<32, TPB, 0, stream>>>(dacc, pcnt, ncnt, psum, nsum, NG);
  k_final<<<1, 1, 0, stream>>>(dacc, (float*)d_out, n);
}